// StratifiedCRPEAttention_420906795538
// MI455X (gfx1250) — compile-verified
//
#include <hip/hip_runtime.h>

typedef float v2f __attribute__((ext_vector_type(2)));
typedef float v4f __attribute__((ext_vector_type(4)));
typedef float v8f __attribute__((ext_vector_type(8)));

namespace {
constexpr int kN     = 16384;    // queries / nodes
constexpr int kM     = 524288;   // pairs
constexpr int kP     = 32;       // pairs per query (M/N), contiguous
constexpr int kHeads = 8;
constexpr int kHdim  = 32;
constexpr int kC     = 256;      // HEADS*HDIM == feature dim
constexpr int kQkv   = 768;      // 3*HEADS*HDIM
constexpr int kBins  = 24;
constexpr float kScale = 0.17677669529663687f;  // 1/sqrt(32)
}  // namespace

// f32 WMMA, D = A(16x4) * B(4x16) + C(16x16), full f32 precision.
__device__ __forceinline__ v8f wmma4(v2f a, v2f b, v8f c) {
  return __builtin_amdgcn_wmma_f32_16x16x4_f32(
      /*neg_a=*/false, a, /*neg_b=*/false, b,
      /*c_mod=*/(short)0, c, /*reuse_a=*/false, /*reuse_b=*/false);
}

// ---------------------------------------------------------------------------
// Kernel A: qkv = feats(16384x256) @ w_qkv^T(256x768) + b_qkv; split q/k/v,
// scale q. One wave per 16x16 output tile; 64 V_WMMA_F32_16X16X4_F32 per tile.
// A-frag layout (ISA 7.12.2, 32-bit A 16x4): lane&15 = M-row, lane>>4 selects
// K-pair {0,1} vs {2,3}; symmetric for B from the row-major weight matrix.
// ---------------------------------------------------------------------------
__global__ void __launch_bounds__(256) qkv_gemm_kernel(
    const float* __restrict__ feats, const float* __restrict__ w_qkv,
    const float* __restrict__ b_qkv, float* __restrict__ q_ws,
    float* __restrict__ k_ws, float* __restrict__ v_ws) {
  const int lane = threadIdx.x & 31;
  const int wave = threadIdx.x >> 5;
  const int tile = blockIdx.x * 8 + wave;
  const int colTiles = kQkv / 16;  // 48
  const int tr = tile / colTiles;
  const int tc = tile % colTiles;
  const int lo = lane & 15;
  const int hi = lane >> 4;

  const float* arow = feats + (size_t)(tr * 16 + lo) * kC + 2 * hi;
  const float* brow = w_qkv + (size_t)(tc * 16 + lo) * kC + 2 * hi;

  v8f acc = {};
#pragma unroll 8
  for (int k = 0; k < kC; k += 4) {
    v2f a = *(const v2f*)(arow + k);
    v2f b = *(const v2f*)(brow + k);
    acc = wmma4(a, b, acc);
  }

  const int gc = tc * 16 + lo;   // global qkv column 0..767 (tile never spans q/k/v)
  const int which = gc >> 8;     // 0=q, 1=k, 2=v
  const int within = gc & 255;
  const float bias = b_qkv[gc];
  const float s = (which == 0) ? kScale : 1.0f;
  float* dst = (which == 0) ? q_ws : (which == 1) ? k_ws : v_ws;
#pragma unroll
  for (int r = 0; r < 8; ++r) {
    const int orow = tr * 16 + r + 8 * hi;  // C/D layout: VGPR r -> M = r + 8*hi
    dst[(size_t)orow * kC + within] = (acc[r] + bias) * s;
  }
}

// ---------------------------------------------------------------------------
// Kernel B: one wave32 per query (its 32 pairs are contiguous).
// Phase 1: lane = pair. b128 gathers of k-row + 6 CRPE tables, 8 head logits.
// Phase 2: segment softmax == 5-step shfl_xor wave reduction per head.
// Phase 3: LDS transpose (attn, ki, bin idx); lane j owns the 8 contiguous
//          output elements [8j, 8j+8) — all inside head h=j>>2, so each pair
//          costs 1 LDS weight + 8 b128 vector gathers (v + 3 v-tables).
// All gathers hit the 192MB L2 (k/v/tables total < 50MB).
// ---------------------------------------------------------------------------
__global__ void __launch_bounds__(256) pair_attn_kernel(
    const float* __restrict__ coords, const int* __restrict__ pair_idx,
    const float* __restrict__ q_ws, const float* __restrict__ k_ws,
    const float* __restrict__ v_ws, const float* __restrict__ q_tab,
    const float* __restrict__ k_tab, const float* __restrict__ v_tab,
    float* __restrict__ agg) {
  const int lane = threadIdx.x & 31;
  const int wave = threadIdx.x >> 5;
  const int qi = blockIdx.x * 8 + wave;
  const int m = qi * kP + lane;
  const int ki = pair_idx[kM + m];  // second row of (2, M) index array

  __shared__ float s_attn[8][kP][kHeads];  // 8 KB
  __shared__ int s_ki[8][kP];
  __shared__ int s_r[8][kP][3];

  // Relative-position bin indices (truncation toward zero matches astype(i32)).
  const float kBinSize = 2.0f * (2.0f * 0.2f) / 24.0f;
  int r0, r1, r2;
  {
    float relx = (coords[qi * 3 + 0] - coords[ki * 3 + 0]) / kBinSize + 12.0f;
    float rely = (coords[qi * 3 + 1] - coords[ki * 3 + 1]) / kBinSize + 12.0f;
    float relz = (coords[qi * 3 + 2] - coords[ki * 3 + 2]) / kBinSize + 12.0f;
    r0 = min(max((int)relx, 0), kBins - 1);
    r1 = min(max((int)rely, 0), kBins - 1);
    r2 = min(max((int)relz, 0), kBins - 1);
  }

  // Phase 1: logits, vectorized as float4 chunks (all rows 16B-aligned).
  const v4f* q4  = (const v4f*)(q_ws + (size_t)qi * kC);  // uniform per wave
  const v4f* k4  = (const v4f*)(k_ws + (size_t)ki * kC);
  const v4f* qt0 = (const v4f*)(q_tab + (size_t)(0 * kBins + r0) * kC);
  const v4f* qt1 = (const v4f*)(q_tab + (size_t)(1 * kBins + r1) * kC);
  const v4f* qt2 = (const v4f*)(q_tab + (size_t)(2 * kBins + r2) * kC);
  const v4f* kt0 = (const v4f*)(k_tab + (size_t)(0 * kBins + r0) * kC);
  const v4f* kt1 = (const v4f*)(k_tab + (size_t)(1 * kBins + r1) * kC);
  const v4f* kt2 = (const v4f*)(k_tab + (size_t)(2 * kBins + r2) * kC);

  float logit[kHeads];
#pragma unroll
  for (int h = 0; h < kHeads; ++h) {
    v4f a4 = {}, b4 = {};
#pragma unroll
    for (int c = 0; c < kHdim / 4; ++c) {
      const int o = h * (kHdim / 4) + c;  // float4 index
      const v4f kg = k4[o];
      const v4f qt = qt0[o] + qt1[o] + qt2[o];
      const v4f kt = kt0[o] + kt1[o] + kt2[o];
      a4 += q4[o] * (kg + qt);
      b4 += kg * kt;
    }
    const v4f t = a4 + b4;
    logit[h] = (t.x + t.y) + (t.z + t.w);
  }

  // Phase 2: segment softmax over the 32 pairs (== lanes of this wave).
#pragma unroll
  for (int h = 0; h < kHeads; ++h) {
    float mx = logit[h];
#pragma unroll
    for (int s = 16; s >= 1; s >>= 1) mx = fmaxf(mx, __shfl_xor(mx, s, 32));
    const float ex = __expf(logit[h] - mx);
    float den = ex;
#pragma unroll
    for (int s = 16; s >= 1; s >>= 1) den += __shfl_xor(den, s, 32);
    s_attn[wave][lane][h] = ex / den;
  }
  s_ki[wave][lane] = ki;
  s_r[wave][lane][0] = r0;
  s_r[wave][lane][1] = r1;
  s_r[wave][lane][2] = r2;
  __syncthreads();

  // Phase 3: lane j owns elements [8j, 8j+8), all within head h = j>>2.
  const int h = lane >> 2;
  const int ob = lane * 8;             // element base (32B aligned)
  const int ob4 = ob / 4;              // float4 base
  v4f acc0 = {}, acc1 = {};
  for (int p = 0; p < kP; ++p) {
    const int kp = s_ki[wave][p];
    const int pr0 = s_r[wave][p][0];
    const int pr1 = s_r[wave][p][1];
    const int pr2 = s_r[wave][p][2];
    const v4f* vr  = (const v4f*)(v_ws + (size_t)kp * kC) + ob4;
    const v4f* vt0 = (const v4f*)(v_tab + (size_t)(0 * kBins + pr0) * kC) + ob4;
    const v4f* vt1 = (const v4f*)(v_tab + (size_t)(1 * kBins + pr1) * kC) + ob4;
    const v4f* vt2 = (const v4f*)(v_tab + (size_t)(2 * kBins + pr2) * kC) + ob4;
    const float w = s_attn[wave][p][h];
    acc0 += w * (vr[0] + vt0[0] + vt1[0] + vt2[0]);
    acc1 += w * (vr[1] + vt0[1] + vt1[1] + vt2[1]);
  }
  v4f* outp = (v4f*)(agg + (size_t)qi * kC + ob);
  outp[0] = acc0;
  outp[1] = acc1;
}

// ---------------------------------------------------------------------------
// Kernel C: out = agg(16384x256) @ w_proj^T(256x256) + b_proj. Same WMMA tiling.
// ---------------------------------------------------------------------------
__global__ void __launch_bounds__(256) proj_gemm_kernel(
    const float* __restrict__ agg, const float* __restrict__ w_proj,
    const float* __restrict__ b_proj, float* __restrict__ out) {
  const int lane = threadIdx.x & 31;
  const int wave = threadIdx.x >> 5;
  const int tile = blockIdx.x * 8 + wave;
  const int colTiles = kC / 16;  // 16
  const int tr = tile / colTiles;
  const int tc = tile % colTiles;
  const int lo = lane & 15;
  const int hi = lane >> 4;

  const float* arow = agg + (size_t)(tr * 16 + lo) * kC + 2 * hi;
  const float* brow = w_proj + (size_t)(tc * 16 + lo) * kC + 2 * hi;

  v8f acc = {};
#pragma unroll 8
  for (int k = 0; k < kC; k += 4) {
    v2f a = *(const v2f*)(arow + k);
    v2f b = *(const v2f*)(brow + k);
    acc = wmma4(a, b, acc);
  }

  const int gc = tc * 16 + lo;
  const float bias = b_proj[gc];
#pragma unroll
  for (int r = 0; r < 8; ++r) {
    const int orow = tr * 16 + r + 8 * hi;
    out[(size_t)orow * kC + gc] = acc[r] + bias;
  }
}

extern "C" void kernel_launch(void* const* d_in, const int* in_sizes, int n_in,
                              void* d_out, int out_size, void* d_ws,
                              size_t ws_size, hipStream_t stream) {
  (void)in_sizes; (void)n_in; (void)out_size; (void)ws_size;
  const float* feats  = (const float*)d_in[0];
  const float* coords = (const float*)d_in[1];
  const int* pair_idx = (const int*)d_in[2];
  const float* w_qkv  = (const float*)d_in[3];
  const float* b_qkv  = (const float*)d_in[4];
  const float* q_tab  = (const float*)d_in[5];
  const float* k_tab  = (const float*)d_in[6];
  const float* v_tab  = (const float*)d_in[7];
  const float* w_proj = (const float*)d_in[8];
  const float* b_proj = (const float*)d_in[9];
  float* out = (float*)d_out;

  // Workspace layout: q | k | v | agg, each N*256 f32 (64 MB total).
  float* q_ws = (float*)d_ws;
  float* k_ws = q_ws + (size_t)kN * kC;
  float* v_ws = k_ws + (size_t)kN * kC;
  float* agg  = v_ws + (size_t)kN * kC;

  // (1024 row-tiles x 48 col-tiles) waves, 8 waves per 256-thread block.
  qkv_gemm_kernel<<<(kN / 16) * (kQkv / 16) / 8, 256, 0, stream>>>(
      feats, w_qkv, b_qkv, q_ws, k_ws, v_ws);
  // One wave per query, 8 queries per block.
  pair_attn_kernel<<<kN / 8, 256, 0, stream>>>(
      coords, pair_idx, q_ws, k_ws, v_ws, q_tab, k_tab, v_tab, agg);
  // (1024 x 16) tiles, 8 waves per block.
  proj_gemm_kernel<<<(kN / 16) * (kC / 16) / 8, 256, 0, stream>>>(
      agg, w_proj, b_proj, out);
}